// EmbeddingLayer_35278861369556
// MI455X (gfx1250) — compile-verified
//
#include <hip/hip_runtime.h>

// Problem constants (match reference)
#define T_  26
#define V_  200000
#define D_  64
#define B_  4096
#define L_  20
#define N_  (B_ * L_)
#define NBAGS (T_ * B_)
#define WAVES_PER_BLOCK 8
#define ROW_BYTES (D_ * 4)         // 256 B per embedding row
#define BAG_BYTES (L_ * ROW_BYTES) // 5120 B staged per bag

#if __has_builtin(__builtin_amdgcn_global_load_async_to_lds_b128)
#define HAVE_ASYNC_LDS 1
#else
#define HAVE_ASYNC_LDS 0
#endif

// 128-bit payload type as the async builtin expects: int __vector(4)
typedef int v4i __attribute__((vector_size(16)));
typedef __attribute__((address_space(1))) v4i* global_v4i_ptr;
typedef __attribute__((address_space(3))) v4i* lds_v4i_ptr;
// Native 2x f32 vector for the nontemporal store (HIP float2 is a struct).
typedef float v2f __attribute__((vector_size(8)));

static __device__ __forceinline__ void wait_async_zero() {
#if __has_builtin(__builtin_amdgcn_s_wait_asynccnt)
    __builtin_amdgcn_s_wait_asynccnt(0);
    asm volatile("" ::: "memory");
#else
    asm volatile("s_wait_asynccnt 0" ::: "memory");
#endif
}

__global__ __launch_bounds__(256) void embbag_sum_kernel(
    const float* __restrict__ W,      // [T, V, D]
    const int*   __restrict__ lS_i,   // [T, N]
    float*       __restrict__ out)    // [T, B, D]
{
#if HAVE_ASYNC_LDS
    __shared__ char smem[WAVES_PER_BLOCK * BAG_BYTES];
#endif
    const int lane = threadIdx.x & 31;
    // Force wave-id into an SGPR so the index loads below become s_load (SMEM).
    const int wv   = __builtin_amdgcn_readfirstlane((int)(threadIdx.x >> 5));
    const int bag  = (int)blockIdx.x * WAVES_PER_BLOCK + wv;   // grid sized exactly

    const int t = bag / B_;
    const int b = bag - t * B_;

    const int*   idxp = lS_i + (size_t)t * N_ + (size_t)b * L_;          // wave-uniform
    const float* Wt   = W + (size_t)t * ((size_t)V_ * (size_t)D_);

    v2f acc = {0.0f, 0.0f};

#if HAVE_ASYNC_LDS
    char* lbuf = smem + wv * BAG_BYTES;

    // Stage the bag's 20 rows into LDS with async DMA loads (ASYNCcnt path).
    // Each b128 async op: lanes 0-15 carry row 2j, lanes 16-31 carry row 2j+1
    // -> 512 B per instruction, 10 instructions per bag, no VGPR staging.
    #pragma unroll
    for (int j = 0; j < L_ / 2; ++j) {
        const int i0 = idxp[2 * j];
        const int i1 = idxp[2 * j + 1];
        const char* g0 = (const char*)(Wt + (size_t)i0 * D_);
        const char* g1 = (const char*)(Wt + (size_t)i1 * D_);
        const char* g  = (lane < 16) ? g0 : g1;
        g += (size_t)(lane & 15) * 16;
        v4i* gp = (v4i*)const_cast<char*>(g);
        v4i* lp = (v4i*)(lbuf + j * (2 * ROW_BYTES) + lane * 16);
        __builtin_amdgcn_global_load_async_to_lds_b128(
            (global_v4i_ptr)gp, (lds_v4i_ptr)lp, 0, 0);
    }

    wait_async_zero();

    // Reduce 20 rows out of LDS: lane reads its float2 column of every row.
    // Addresses r*256 + lane*8 are bank-conflict free across the wave.
    const v2f* rows = (const v2f*)lbuf;
    #pragma unroll
    for (int r = 0; r < L_; ++r) {
        acc += rows[r * (D_ / 2) + lane];
    }
#else
    // Direct gather path: idx is in an SGPR, so each row read is a
    // saddr-form global_load_b64 with voffset = lane*8 (one full 256B row
    // per wave per instruction). 20 independent loads -> deep MLP.
    #pragma unroll
    for (int l = 0; l < L_; ++l) {
        const int idx = idxp[l];
        const v2f* row = (const v2f*)(Wt + (size_t)idx * D_);
        acc += row[lane];
    }
#endif

    // Coalesced 256 B store per wave. Non-temporal: output is write-once and
    // never re-read, so don't let it displace L2-resident embedding rows
    // (per-table unique gather set ~17 MB fits in the 192 MB L2).
    v2f* o = (v2f*)(out + (size_t)bag * D_);
    __builtin_nontemporal_store(acc, &o[lane]);
}

extern "C" void kernel_launch(void* const* d_in, const int* in_sizes, int n_in,
                              void* d_out, int out_size, void* d_ws, size_t ws_size,
                              hipStream_t stream) {
    (void)in_sizes; (void)n_in; (void)out_size; (void)d_ws; (void)ws_size;
    const float* W    = (const float*)d_in[0];  // weights [T,V,D] fp32
    // d_in[1] = lS_o (offsets) -- implied by fixed bag length L, unused
    const int*   lS_i = (const int*)d_in[2];    // [T, N] int32
    float*       out  = (float*)d_out;          // [T, B, D] fp32

    const int blocks = NBAGS / WAVES_PER_BLOCK; // 106496 / 8 = 13312, exact
    hipLaunchKernelGGL(embbag_sum_kernel, dim3(blocks), dim3(256), 0, stream,
                       W, lS_i, out);
}